// DeformAttn_67456756351201
// MI455X (gfx1250) — compile-verified
//
#include <hip/hip_runtime.h>
#include <math.h>

#define DIMC 48
#define Hh 128
#define Ww 128
#define HID 288
#define C2 96
#define NPIX (Hh * Ww)
#define K2LEN (HID * 9) /* 2592 */

typedef __attribute__((ext_vector_type(2))) float v2f;
typedef __attribute__((ext_vector_type(8))) float v8f;

// ---------------------------------------------------------------------------
// CDNA5 async global->LDS copies (ASYNCcnt-tracked), GV addressing form.
// vdst = VGPR holding LDS byte offset, vaddr = 64-bit global address.
// ---------------------------------------------------------------------------
__device__ __forceinline__ void async_copy_b32(unsigned lds_off, const void* gaddr) {
  asm volatile("global_load_async_to_lds_b32 %0, %1, off"
               :: "v"(lds_off), "v"(gaddr)
               : "memory");
}

__device__ __forceinline__ void async_copy_b128(unsigned lds_off, const void* gaddr) {
  asm volatile("global_load_async_to_lds_b128 %0, %1, off"
               :: "v"(lds_off), "v"(gaddr)
               : "memory");
}

__device__ __forceinline__ void wait_async0() {
  asm volatile("s_wait_asynccnt 0x0" ::: "memory");
}

__device__ __forceinline__ unsigned lds_offset(const void* p) {
  return (unsigned)(unsigned long long)p;
}

// ---------------------------------------------------------------------------
// bilinear sampling helpers (zero outside image, matching the reference's
// "clip then multiply by validity" semantics)
// ---------------------------------------------------------------------------
__device__ __forceinline__ float sample_corner(const float* __restrict__ ch, int y, int x) {
  bool valid = (y >= 0) && (y < Hh) && (x >= 0) && (x < Ww);
  int yc = min(max(y, 0), Hh - 1);
  int xc = min(max(x, 0), Ww - 1);
  return valid ? ch[yc * Ww + xc] : 0.0f;
}

__device__ __forceinline__ float bilinear(const float* __restrict__ ch, float ys, float xs) {
  float y0f = floorf(ys), x0f = floorf(xs);
  float wy = ys - y0f, wx = xs - x0f;
  int y0 = (int)y0f, x0 = (int)x0f;
  float v00 = sample_corner(ch, y0,     x0    );
  float v01 = sample_corner(ch, y0,     x0 + 1);
  float v10 = sample_corner(ch, y0 + 1, x0    );
  float v11 = sample_corner(ch, y0 + 1, x0 + 1);
  return v00 * (1.0f - wy) * (1.0f - wx) + v01 * (1.0f - wy) * wx +
         v10 * wy * (1.0f - wx)          + v11 * wy * wx;
}

// ---------------------------------------------------------------------------
// Kernel 1: deformable 1x1 conv, 48 -> 288, via WMMA f32 16x16x4 GEMM
// block = 288 threads (9 waves), each block handles 16 consecutive pixels
// ---------------------------------------------------------------------------
__global__ __launch_bounds__(288)
void k1_stage1(const float* __restrict__ x, const float* __restrict__ off_w1,
               const float* __restrict__ off_b1, const float* __restrict__ w1,
               const float* __restrict__ b1, float* __restrict__ hidden) {
  __shared__ float s_tile[DIMC][16];
  __shared__ float s_ys[16];
  __shared__ float s_xs[16];
  int tid = threadIdx.x;
  int pix0 = blockIdx.x * 16;

  // offsets for the 16 pixels (2-channel 1x1 conv)
  if (tid < 16) {
    int pix = pix0 + tid;
    float oy = off_b1[0], ox = off_b1[1];
    for (int c = 0; c < DIMC; ++c) {
      float xv = x[c * NPIX + pix];
      oy += off_w1[c] * xv;          // off_w1[0][c]
      ox += off_w1[DIMC + c] * xv;   // off_w1[1][c]
    }
    s_ys[tid] = (float)(pix / Ww) + oy;
    s_xs[tid] = (float)(pix % Ww) + ox;
  }
  __syncthreads();

  // bilinear gather of the 48x16 sampled activation tile into LDS
  for (int idx = tid; idx < DIMC * 16; idx += 288) {
    int c = idx >> 4, p = idx & 15;
    s_tile[c][p] = bilinear(x + c * NPIX, s_ys[p], s_xs[p]);
  }
  __syncthreads();

  // WMMA GEMM: hidden[m, pix] = W1[m, :] . s[:, pix] + b1[m]
  int wave = tid >> 5;
  int lane = tid & 31;
  int mlane = lane & 15;
  int hi = lane >> 4; // 0: K k,k+1, 1: K k+2,k+3 (per 16x4 A layout)
  for (int t = 0; t < 2; ++t) {
    int m0 = (wave * 2 + t) * 16;
    v8f acc = {};
    for (int k = 0; k < DIMC; k += 4) {
      v2f a, b;
      a.x = w1[(m0 + mlane) * DIMC + k + 2 * hi];
      a.y = w1[(m0 + mlane) * DIMC + k + 2 * hi + 1];
      b.x = s_tile[k + 2 * hi][mlane];
      b.y = s_tile[k + 2 * hi + 1][mlane];
      acc = __builtin_amdgcn_wmma_f32_16x16x4_f32(false, a, false, b,
                                                  (short)0, acc, false, false);
    }
    for (int r = 0; r < 8; ++r) {
      int m = m0 + r + hi * 8;
      hidden[m * NPIX + pix0 + mlane] = acc[r] + b1[m];
    }
  }
}

// ---------------------------------------------------------------------------
// Kernel 2: 3x3 offset conv on hidden: im2col GEMM M=18 (padded 32),
// K=288*9=2592, N=16 pixels per block. 64 threads (2 waves), hidden patch
// staged into 62KB LDS via async global->LDS copies, then 648
// v_wmma_f32_16x16x4_f32 per wave.
// ---------------------------------------------------------------------------
__global__ __launch_bounds__(64)
void k2_offconv(const float* __restrict__ hidden, const float* __restrict__ off_w2,
                const float* __restrict__ off_b2, float* __restrict__ off2) {
  __shared__ float hl[HID * 3 * 18]; // [c][dy 0..2][dx 0..17], cols x0-1 .. x0+16
  int tid = threadIdx.x;
  int pix0 = blockIdx.x * 16;
  int h = pix0 / Ww;
  int wcol = pix0 % Ww;

  // async staging: memory -> LDS directly (no VGPR round-trip); zero-fill pad
  for (int idx = tid; idx < HID * 3 * 18; idx += 64) {
    int c = idx / 54;
    int rem = idx % 54;
    int dy = rem / 18;
    int dx = rem % 18;
    int yy = h + dy - 1;
    int xx = wcol + dx - 1;
    if (yy >= 0 && yy < Hh && xx >= 0 && xx < Ww) {
      async_copy_b32(lds_offset(&hl[idx]), &hidden[c * NPIX + yy * Ww + xx]);
    } else {
      hl[idx] = 0.0f; // ds_store zero-fill for padding
    }
  }
  wait_async0();
  __syncthreads();

  int wave = tid >> 5, lane = tid & 31;
  int mlane = lane & 15, hi = lane >> 4;
  int m0 = wave * 16;
  int mrow = m0 + mlane;
  bool mvalid = (mrow < 18);
  const float* wrow = off_w2 + (mvalid ? mrow : 0) * K2LEN;
  if (mvalid) __builtin_prefetch(wrow, 0, 1); // -> global_prefetch_b8

  v8f acc = {};
  for (int k = 0; k < K2LEN; k += 4) {
    int ka = k + 2 * hi;
    v2f a, b;
    a.x = mvalid ? wrow[ka] : 0.0f;
    a.y = mvalid ? wrow[ka + 1] : 0.0f;
    int c0 = ka / 9, t0 = ka % 9;
    b.x = hl[c0 * 54 + (t0 / 3) * 18 + (t0 % 3) + mlane];
    int kb = ka + 1;
    int c1 = kb / 9, t1 = kb % 9;
    b.y = hl[c1 * 54 + (t1 / 3) * 18 + (t1 % 3) + mlane];
    acc = __builtin_amdgcn_wmma_f32_16x16x4_f32(false, a, false, b,
                                                (short)0, acc, false, false);
  }
  for (int r = 0; r < 8; ++r) {
    int m = m0 + r + hi * 8;
    if (m < 18)
      off2[m * NPIX + pix0 + mlane] = acc[r] + off_b2[m];
  }
}

// ---------------------------------------------------------------------------
// Kernel 3: deformable depthwise 3x3: bilinear gather + 9-tap MAC
// one thread per (channel, pixel); pixels contiguous within a wave
// ---------------------------------------------------------------------------
__global__ __launch_bounds__(256)
void k3_depthwise(const float* __restrict__ hidden, const float* __restrict__ off2,
                  const float* __restrict__ w2, const float* __restrict__ b2,
                  float* __restrict__ qkv) {
  int idx = blockIdx.x * 256 + threadIdx.x;
  if (idx >= HID * NPIX) return;
  int pix = idx & (NPIX - 1);
  int c = idx / NPIX;
  int h = pix / Ww, w = pix % Ww;
  const float* ch = hidden + c * NPIX;
  float acc = 0.0f;
#pragma unroll
  for (int tap = 0; tap < 9; ++tap) {
    float dy = off2[(2 * tap) * NPIX + pix];
    float dx = off2[(2 * tap + 1) * NPIX + pix];
    float ys = (float)h - 1.0f + (float)(tap / 3) + dy;
    float xs = (float)w - 1.0f + (float)(tap % 3) + dx;
    acc += w2[c * 9 + tap] * bilinear(ch, ys, xs);
  }
  qkv[idx] = acc + b2[c];
}

// ---------------------------------------------------------------------------
// Kernel 4: per-8x8-patch circular conv (== irfft2(rfft2(q)*rfft2(k))) +
// per-pixel LayerNorm over 96 channels + v*o. One block per patch.
// q/k chunks staged with async b128 copies (8-float contiguous patch rows).
// LDS: 24KB (o) + 2 x 12KB (q/k chunk) = 48KB
// ---------------------------------------------------------------------------
#define CHUNK 48
__global__ __launch_bounds__(256)
void k4_patch_attn(const float* __restrict__ qkv, const float* __restrict__ ln_w,
                   const float* __restrict__ ln_b, float* __restrict__ vo) {
  __shared__ float ol[C2 * 64];
  __shared__ float ql[CHUNK * 64];
  __shared__ float kl[CHUNK * 64];
  int tid = threadIdx.x;
  int pt = blockIdx.x;
  int pr = pt >> 4, pc = pt & 15;

  for (int cb = 0; cb < C2; cb += CHUNK) {
    // async staging: each transfer moves 4 contiguous floats; a patch row
    // (u fixed) is 8 contiguous floats in global and in LDS -> 2 x b128.
    // total transfers: CHUNK * 8 rows * 2 halves * 2 (q,k) = 1536
    for (int idx = tid; idx < CHUNK * 16 * 2; idx += 256) {
      int isk = idx >= CHUNK * 16;
      int rem = isk ? (idx - CHUNK * 16) : idx;
      int c = rem >> 4;           // channel within chunk
      int sub = rem & 15;         // 8 rows x 2 halves
      int u = sub >> 1;
      int half = sub & 1;
      int gpix = (pr * 8 + u) * Ww + pc * 8 + half * 4;
      const float* src = qkv + (isk ? (C2 + cb + c) : (cb + c)) * NPIX + gpix;
      float* dst = (isk ? kl : ql) + c * 64 + u * 8 + half * 4;
      async_copy_b128(lds_offset(dst), src);
    }
    wait_async0();
    __syncthreads();

    for (int idx = tid; idx < CHUNK * 64; idx += 256) {
      int c = idx >> 6, uv = idx & 63;
      int u = uv >> 3, v = uv & 7;
      float acc = 0.0f;
#pragma unroll
      for (int a = 0; a < 8; ++a) {
        int ru = ((u - a) & 7) << 3;
#pragma unroll
        for (int b = 0; b < 8; ++b)
          acc += ql[(c << 6) + (a << 3) + b] * kl[(c << 6) + ru + ((v - b) & 7)];
      }
      ol[(cb + c) * 64 + uv] = acc;
    }
    __syncthreads();
  }

  // LayerNorm across channels at each of the 64 patch pixels
  if (tid < 64) {
    float mu = 0.0f, sq = 0.0f;
    for (int c = 0; c < C2; ++c) {
      float o = ol[c * 64 + tid];
      mu += o;
      sq += o * o;
    }
    mu *= (1.0f / C2);
    float var = sq * (1.0f / C2) - mu * mu;
    float rs = rsqrtf(var + 1e-5f);
    for (int c = 0; c < C2; ++c)
      ol[c * 64 + tid] = (ol[c * 64 + tid] - mu) * rs * ln_w[c] + ln_b[c];
  }
  __syncthreads();

  for (int idx = tid; idx < C2 * 64; idx += 256) {
    int c = idx >> 6, uv = idx & 63;
    int pix = (pr * 8 + (uv >> 3)) * Ww + pc * 8 + (uv & 7);
    vo[c * NPIX + pix] = qkv[(2 * C2 + c) * NPIX + pix] * ol[c * 64 + uv];
  }
}

// ---------------------------------------------------------------------------
// Kernel 5: deformable 1x1 conv, 96 -> 48 (same structure as kernel 1)
// block = 96 threads (3 waves), 16 pixels per block
// ---------------------------------------------------------------------------
__global__ __launch_bounds__(96)
void k5_stage3(const float* __restrict__ vo, const float* __restrict__ off_w3,
               const float* __restrict__ off_b3, const float* __restrict__ w3,
               const float* __restrict__ b3, float* __restrict__ out) {
  __shared__ float s_tile[C2][16];
  __shared__ float s_ys[16];
  __shared__ float s_xs[16];
  int tid = threadIdx.x;
  int pix0 = blockIdx.x * 16;

  if (tid < 16) {
    int pix = pix0 + tid;
    float oy = off_b3[0], ox = off_b3[1];
    for (int c = 0; c < C2; ++c) {
      float xv = vo[c * NPIX + pix];
      oy += off_w3[c] * xv;
      ox += off_w3[C2 + c] * xv;
    }
    s_ys[tid] = (float)(pix / Ww) + oy;
    s_xs[tid] = (float)(pix % Ww) + ox;
  }
  __syncthreads();

  for (int idx = tid; idx < C2 * 16; idx += 96) {
    int c = idx >> 4, p = idx & 15;
    s_tile[c][p] = bilinear(vo + c * NPIX, s_ys[p], s_xs[p]);
  }
  __syncthreads();

  int wave = tid >> 5, lane = tid & 31;
  int mlane = lane & 15, hi = lane >> 4;
  int m0 = wave * 16;
  v8f acc = {};
  for (int k = 0; k < C2; k += 4) {
    v2f a, b;
    a.x = w3[(m0 + mlane) * C2 + k + 2 * hi];
    a.y = w3[(m0 + mlane) * C2 + k + 2 * hi + 1];
    b.x = s_tile[k + 2 * hi][mlane];
    b.y = s_tile[k + 2 * hi + 1][mlane];
    acc = __builtin_amdgcn_wmma_f32_16x16x4_f32(false, a, false, b,
                                                (short)0, acc, false, false);
  }
  for (int r = 0; r < 8; ++r) {
    int m = m0 + r + hi * 8;
    out[m * NPIX + pix0 + mlane] = acc[r] + b3[m];
  }
}

// ---------------------------------------------------------------------------
extern "C" void kernel_launch(void* const* d_in, const int* in_sizes, int n_in,
                              void* d_out, int out_size, void* d_ws, size_t ws_size,
                              hipStream_t stream) {
  (void)in_sizes; (void)n_in; (void)out_size; (void)ws_size;
  const float* x      = (const float*)d_in[0];
  const float* off_w1 = (const float*)d_in[1];
  const float* off_b1 = (const float*)d_in[2];
  const float* w1     = (const float*)d_in[3];
  const float* b1     = (const float*)d_in[4];
  const float* off_w2 = (const float*)d_in[5];
  const float* off_b2 = (const float*)d_in[6];
  const float* w2     = (const float*)d_in[7];
  const float* b2     = (const float*)d_in[8];
  const float* ln_w   = (const float*)d_in[9];
  const float* ln_b   = (const float*)d_in[10];
  const float* off_w3 = (const float*)d_in[11];
  const float* off_b3 = (const float*)d_in[12];
  const float* w3     = (const float*)d_in[13];
  const float* b3     = (const float*)d_in[14];
  float* out = (float*)d_out;

  float* ws     = (float*)d_ws;
  float* hidden = ws;                    // 288 * 16384
  float* off2   = hidden + HID * NPIX;   //  18 * 16384
  float* qkv    = off2 + 18 * NPIX;      // 288 * 16384
  float* vo     = qkv + HID * NPIX;      //  96 * 16384

  hipLaunchKernelGGL(k1_stage1, dim3(NPIX / 16), dim3(288), 0, stream,
                     x, off_w1, off_b1, w1, b1, hidden);
  hipLaunchKernelGGL(k2_offconv, dim3(NPIX / 16), dim3(64), 0, stream,
                     hidden, off_w2, off_b2, off2);
  hipLaunchKernelGGL(k3_depthwise, dim3((HID * NPIX) / 256), dim3(256), 0, stream,
                     hidden, off2, w2, b2, qkv);
  hipLaunchKernelGGL(k4_patch_attn, dim3(256), dim3(256), 0, stream,
                     qkv, ln_w, ln_b, vo);
  hipLaunchKernelGGL(k5_stage3, dim3(NPIX / 16), dim3(96), 0, stream,
                     vo, off_w3, off_b3, w3, b3, out);
}